// LNN_42923903156922
// MI455X (gfx1250) — compile-verified
//
#include <hip/hip_runtime.h>

// ============================================================================
// LNN dynamics on MI455X (gfx1250), f32 end-to-end via V_WMMA_F32_16X16X4_F32.
//
// Math (exact, since only elementwise silu has curvature):
//   forward : u_k = h_{k-1} W_k + b_k, h_k = silu(u_k), k=0..3, h_{-1}=z
//   adjoint : d3 = w4 ; c_k = d_k * silu'(u_k) ; d_{k-1} = c_k W_k^T
//   grad    : g = c_0 W_0^T                               (BSx32)
//   Hessian : H = sum_k Ju_k^T diag(s_k) Ju_k, s_k = d_k * silu''(u_k)
//             Ju_0 = W0^T ; Ju_{k+1} = W_{k+1}^T diag(silu'(u_k)) Ju_k
//   + 0.2*I added to H[16:,16:] (from 0.1*p.p), then 16x16 solve.
//
// Workspace layout (floats), required ws_size >= ~136 MB:
//   U[4][BS][256], Hact[4][BS][256], C[4][BS][256], S[4][BS][256], G[BS][32]
// Per-sample Jacobians (32KB each) never touch HBM: kept in LDS (64KB
// ping-pong per wave, 4 waves/WG = 256KB of the 320KB WGP LDS). Weights
// (<1MB total) are L2-resident, so per-wave streaming loads are cheap.
// ============================================================================

typedef __attribute__((ext_vector_type(2))) float v2f;
typedef __attribute__((ext_vector_type(8))) float v8f;

#define LNN_BS   8192
#define LNN_HID  256
#define LNN_D    16
#define LNN_DZ   32

__device__ __forceinline__ v8f v8zero() {
  v8f z;
#pragma unroll
  for (int i = 0; i < 8; ++i) z[i] = 0.0f;
  return z;
}

// D(16x16,f32) = A(16x4,f32) * B(4x16,f32) + C  — CDNA5 wave32 WMMA
__device__ __forceinline__ v8f wmma4(v2f a, v2f b, v8f c) {
  return __builtin_amdgcn_wmma_f32_16x16x4_f32(
      /*neg_a=*/false, a, /*neg_b=*/false, b,
      /*c_mod=*/(short)0, c, /*reuse_a=*/false, /*reuse_b=*/false);
}

__device__ __forceinline__ float sigm(float x) { return 1.0f / (1.0f + __expf(-x)); }
__device__ __forceinline__ float silu_f (float u) { return u * sigm(u); }
__device__ __forceinline__ float silu_d1(float u) { float s = sigm(u); return s * (1.0f + u * (1.0f - s)); }
__device__ __forceinline__ float silu_d2(float u) { float s = sigm(u); return s * (1.0f - s) * (2.0f + u * (1.0f - 2.0f * s)); }

// ----------------------------------------------------------------------------
// Batched WMMA GEMM: Y = A(MxK) * Bop(KxN). One 32x32 output block per wave
// (2x2 tiles of 16x16, 4 v8f accumulators): per K-step 2 A-frags + 2 B-frags
// feed 4 WMMAs. TRANSB/EPI are compile-time so the K-loop is branch-free.
// TRANSB: Bop[k][n] = B[n*K + k] (i.e. W^T with W stored row-major (N,K)).
// EPI 0: out0 = Y + bias (=U), out1 = silu(U)                (forward layer)
// EPI 1: d = Y; out0 = d*silu'(Uref), out1 = d*silu''(Uref)  (backward)
// EPI 2: out0 = Y                                            (gradient)
// ----------------------------------------------------------------------------
template <int TRANSB, int EPI>
__global__ __launch_bounds__(256) void lnn_gemm(
    const float* __restrict__ A, const float* __restrict__ B,
    const float* __restrict__ bias, const float* __restrict__ Uref,
    float* __restrict__ out0, float* __restrict__ out1,
    int M, int N, int K)
{
  const int wave = threadIdx.x >> 5;
  const int lane = threadIdx.x & 31;
  const int half = lane >> 4;     // lanes 16..31
  const int lm   = lane & 15;
  const int ngrp = N >> 5;
  const int mgrp = M >> 5;
  const int grp  = blockIdx.x * (blockDim.x >> 5) + wave;
  if (grp >= mgrp * ngrp) return;
  const int m0 = (grp / ngrp) << 5;
  const int n0 = (grp % ngrp) << 5;

  v8f c00 = v8zero(), c01 = v8zero(), c10 = v8zero(), c11 = v8zero();

#pragma unroll 4
  for (int k0 = 0; k0 < K; k0 += 4) {
    const int ka = k0 + 2 * half;              // A/B frag K rows per ISA layout
    // A-frags for m-tiles 0,1 (contiguous float2 along K -> b64 loads)
    v2f a0, a1, b0, b1;
    a0.x = A[(size_t)(m0 + lm) * K + ka];
    a0.y = A[(size_t)(m0 + lm) * K + ka + 1];
    a1.x = A[(size_t)(m0 + 16 + lm) * K + ka];
    a1.y = A[(size_t)(m0 + 16 + lm) * K + ka + 1];
    __builtin_prefetch(&A[(size_t)(m0 + lm) * K + ka + 32], 0, 1);   // global_prefetch_b8
    if (TRANSB == 0) {
      b0.x = B[(size_t)ka * N + n0 + lm];
      b0.y = B[(size_t)(ka + 1) * N + n0 + lm];
      b1.x = B[(size_t)ka * N + n0 + 16 + lm];
      b1.y = B[(size_t)(ka + 1) * N + n0 + 16 + lm];
    } else {
      b0.x = B[(size_t)(n0 + lm) * K + ka];
      b0.y = B[(size_t)(n0 + lm) * K + ka + 1];
      b1.x = B[(size_t)(n0 + 16 + lm) * K + ka];
      b1.y = B[(size_t)(n0 + 16 + lm) * K + ka + 1];
    }
    c00 = wmma4(a0, b0, c00);
    c01 = wmma4(a0, b1, c01);
    c10 = wmma4(a1, b0, c10);
    c11 = wmma4(a1, b1, c11);
  }

  v8f acc[2][2] = { { c00, c01 }, { c10, c11 } };
#pragma unroll
  for (int ti = 0; ti < 2; ++ti) {
#pragma unroll
    for (int tj = 0; tj < 2; ++tj) {
      const int n = n0 + tj * 16 + lm;
#pragma unroll
      for (int r = 0; r < 8; ++r) {
        const int m = m0 + ti * 16 + r + 8 * half;   // C/D VGPR layout
        const size_t idx = (size_t)m * N + n;
        const float y = acc[ti][tj][r];
        if (EPI == 0) {
          const float u = y + bias[n];
          out0[idx] = u;
          out1[idx] = silu_f(u);
        } else if (EPI == 1) {
          const float u = Uref[idx];
          out0[idx] = y * silu_d1(u);
          out1[idx] = y * silu_d2(u);
        } else {
          out0[idx] = y;
        }
      }
    }
  }
}

// Backward seed: d3 = w4 (shared across batch); c3 = w4*silu'(u3), s3 = w4*silu''(u3)
__global__ __launch_bounds__(256) void lnn_back_init(
    const float* __restrict__ U3, const float* __restrict__ W4,
    float* __restrict__ C3, float* __restrict__ S3, int total)
{
  const int i = blockIdx.x * blockDim.x + threadIdx.x;
  if (i >= total) return;
  const float w = W4[i & (LNN_HID - 1)];       // W4 is (256,1)
  const float u = U3[i];
  C3[i] = w * silu_d1(u);
  S3[i] = w * silu_d2(u);
}

// ----------------------------------------------------------------------------
// Fused Jacobian propagation + Hessian accumulation + 16x16 solve.
// One sample per wave; 4 waves/WG. Ju (256x32 f32) ping-pongs in LDS.
//   per k: Hacc += Ju^T diag(s_k) Ju   (2x2 tiles of 16x16, K=256)
//          Ju'   = W_{k+1}^T diag(a'_k) Ju  (16x2 tiles, K=256)
// Then rhs = g[:16] - H[16:,:16] v ; (H[16:,16:]+0.2I) a = rhs ; out=[v,a].
// Dynamic LDS: 4*(2*8192) + 4*512 + 4*(32*33) floats = 287232 bytes.
// ----------------------------------------------------------------------------
__global__ __launch_bounds__(128) void lnn_jac_hess(
    const float* __restrict__ z,
    const float* __restrict__ W0, const float* __restrict__ W1,
    const float* __restrict__ W2, const float* __restrict__ W3,
    const float* __restrict__ Ubuf, const float* __restrict__ Sbuf,
    const float* __restrict__ G, float* __restrict__ out)
{
  extern __shared__ float smem[];
  const int wave = threadIdx.x >> 5;
  const int lane = threadIdx.x & 31;
  const int half = lane >> 4;
  const int lm   = lane & 15;
  const int b = blockIdx.x * 4 + wave;

  float* J0 = smem + wave * (2 * 8192);
  float* J1 = J0 + 8192;
  float* ap = smem + 4 * (2 * 8192) + wave * 512;  // silu'(u_k)
  float* sv = ap + 256;                            // s_k
  float* hs = smem + 4 * (2 * 8192) + 4 * 512 + wave * (32 * 33);

  // Ju_0 = W0^T : J[i*32+d] = W0[d*256+i]
  for (int idx = lane; idx < 8192; idx += 32) {
    const int i = idx >> 5, d = idx & 31;
    J0[idx] = W0[d * LNN_HID + i];
  }
  float* Jc = J0;
  float* Jn = J1;
  v8f hc[2][2];
  hc[0][0] = v8zero(); hc[0][1] = v8zero();
  hc[1][0] = v8zero(); hc[1][1] = v8zero();
  __syncthreads();

  for (int k = 0; k < 4; ++k) {
    // layer-k elementwise tables
    for (int i = lane; i < 256; i += 32) {
      const size_t o = ((size_t)k * LNN_BS + b) * LNN_HID + i;
      ap[i] = silu_d1(Ubuf[o]);
      sv[i] = Sbuf[o];
    }
    __syncthreads();

    // Hacc += (diag(s) Ju)^T * Ju   — M=N=32, K=256
#pragma unroll 2
    for (int k0 = 0; k0 < 256; k0 += 4) {
      const int ka = k0 + 2 * half;
      const float s0 = sv[ka], s1 = sv[ka + 1];
      v2f a0, a1, b0, b1;
      b0.x = Jc[ka * 32 + lm];        b0.y = Jc[(ka + 1) * 32 + lm];
      b1.x = Jc[ka * 32 + 16 + lm];   b1.y = Jc[(ka + 1) * 32 + 16 + lm];
      a0.x = s0 * b0.x;               a0.y = s1 * b0.y;
      a1.x = s0 * b1.x;               a1.y = s1 * b1.y;
      hc[0][0] = wmma4(a0, b0, hc[0][0]);
      hc[0][1] = wmma4(a0, b1, hc[0][1]);
      hc[1][0] = wmma4(a1, b0, hc[1][0]);
      hc[1][1] = wmma4(a1, b1, hc[1][1]);
    }

    // Ju' = W_{k+1}^T diag(a') Ju  — M=256, N=32, K=256
    if (k < 3) {
      const float* W = (k == 0) ? W1 : ((k == 1) ? W2 : W3);
      for (int mt = 0; mt < 16; ++mt) {
        v8f acc0 = v8zero(), acc1 = v8zero();
#pragma unroll 2
        for (int k0 = 0; k0 < 256; k0 += 4) {
          const int ka = k0 + 2 * half;
          v2f a, bb0, bb1;
          a.x  = W[(size_t)ka * LNN_HID + mt * 16 + lm];        // W^T[m][k]
          a.y  = W[(size_t)(ka + 1) * LNN_HID + mt * 16 + lm];
          const float p0 = ap[ka], p1 = ap[ka + 1];
          bb0.x = p0 * Jc[ka * 32 + lm];
          bb0.y = p1 * Jc[(ka + 1) * 32 + lm];
          bb1.x = p0 * Jc[ka * 32 + 16 + lm];
          bb1.y = p1 * Jc[(ka + 1) * 32 + 16 + lm];
          acc0 = wmma4(a, bb0, acc0);
          acc1 = wmma4(a, bb1, acc1);
        }
#pragma unroll
        for (int r = 0; r < 8; ++r) {
          Jn[(mt * 16 + r + 8 * half) * 32 + lm]      = acc0[r];
          Jn[(mt * 16 + r + 8 * half) * 32 + 16 + lm] = acc1[r];
        }
      }
      float* t = Jc; Jc = Jn; Jn = t;
    }
    __syncthreads();
  }

  // spill Hacc (32x32) to LDS, padded to 33 cols; col 32 holds rhs/solution
#pragma unroll
  for (int dt = 0; dt < 2; ++dt)
#pragma unroll
    for (int et = 0; et < 2; ++et)
#pragma unroll
      for (int r = 0; r < 8; ++r)
        hs[(dt * 16 + r + 8 * half) * 33 + et * 16 + lm] = hc[dt][et][r];
  __syncthreads();

  // rhs = g[:16] - H[16:,:16] v ; add 0.2 to diag of H[16:,16:]
  if (lane < 16) {
    float rhs = G[(size_t)b * 32 + lane];
    for (int c2 = 0; c2 < 16; ++c2)
      rhs -= hs[(16 + lane) * 33 + c2] * z[(size_t)b * 32 + 16 + c2];
    hs[(16 + lane) * 33 + 32] = rhs;
    hs[(16 + lane) * 33 + 16 + lane] += 0.2f;
  }
  __syncthreads();

  // Gaussian elimination on the augmented 16x17 system (lane r owns row r)
  for (int p = 0; p < 16; ++p) {
    if (lane < 16 && lane > p) {
      const float f = hs[(16 + lane) * 33 + 16 + p] / hs[(16 + p) * 33 + 16 + p];
      for (int c2 = p; c2 <= 16; ++c2)           // cols 16+p .. 32 (incl. rhs)
        hs[(16 + lane) * 33 + 16 + c2] -= f * hs[(16 + p) * 33 + 16 + c2];
    }
    __syncthreads();
  }
  // back substitution
  for (int p = 15; p >= 0; --p) {
    if (lane == p) {
      float x = hs[(16 + p) * 33 + 32];
      for (int c2 = p + 1; c2 < 16; ++c2)
        x -= hs[(16 + p) * 33 + 16 + c2] * hs[(16 + c2) * 33 + 32];
      hs[(16 + p) * 33 + 32] = x / hs[(16 + p) * 33 + 16 + p];
    }
    __syncthreads();
  }

  if (lane < 16) {
    out[(size_t)b * 32 + lane]      = z[(size_t)b * 32 + 16 + lane];  // v
    out[(size_t)b * 32 + 16 + lane] = hs[(16 + lane) * 33 + 32];      // a
  }
}

// ============================================================================
extern "C" void kernel_launch(void* const* d_in, const int* in_sizes, int n_in,
                              void* d_out, int out_size, void* d_ws, size_t ws_size,
                              hipStream_t stream) {
  (void)in_sizes; (void)n_in; (void)out_size; (void)ws_size;
  // inputs: t, z, W0,b0, W1,b1, W2,b2, W3,b3, W4,b4
  const float* z  = (const float*)d_in[1];
  const float* W[4] = { (const float*)d_in[2], (const float*)d_in[4],
                        (const float*)d_in[6], (const float*)d_in[8] };
  const float* bv[4] = { (const float*)d_in[3], (const float*)d_in[5],
                         (const float*)d_in[7], (const float*)d_in[9] };
  const float* W4 = (const float*)d_in[10];
  float* out = (float*)d_out;

  float* ws = (float*)d_ws;
  const size_t L = (size_t)LNN_BS * LNN_HID;   // 2,097,152 floats per layer slab
  float* U  = ws;                // [4][BS][256] pre-activations
  float* Ha = U  + 4 * L;        // [4][BS][256] silu(U)
  float* C  = Ha + 4 * L;        // [4][BS][256] d_k * silu'(u_k)
  float* S  = C  + 4 * L;        // [4][BS][256] d_k * silu''(u_k)
  float* G  = S  + 4 * L;        // [BS][32] gradient wrt z

  const dim3 blk(256);
  const int grp_full = (LNN_BS / 32) * (LNN_HID / 32);   // 2048 -> 256 blocks
  const int grp_g    = (LNN_BS / 32) * (LNN_DZ / 32);    // 256  -> 32 blocks

  // ---- forward: U_k, h_k ----
  lnn_gemm<0, 0><<<grp_full / 8, blk, 0, stream>>>(z, W[0], bv[0], nullptr,
                                                   U, Ha, LNN_BS, LNN_HID, LNN_DZ);
  for (int k = 1; k < 4; ++k)
    lnn_gemm<0, 0><<<grp_full / 8, blk, 0, stream>>>(Ha + (k - 1) * L, W[k], bv[k], nullptr,
                                                     U + k * L, Ha + k * L,
                                                     LNN_BS, LNN_HID, LNN_HID);

  // ---- backward: c_k, s_k, then g ----
  lnn_back_init<<<(LNN_BS * LNN_HID) / 256, blk, 0, stream>>>(U + 3 * L, W4,
                                                              C + 3 * L, S + 3 * L,
                                                              LNN_BS * LNN_HID);
  for (int k = 3; k >= 1; --k)
    lnn_gemm<1, 1><<<grp_full / 8, blk, 0, stream>>>(C + k * L, W[k], nullptr, U + (k - 1) * L,
                                                     C + (k - 1) * L, S + (k - 1) * L,
                                                     LNN_BS, LNN_HID, LNN_HID);
  lnn_gemm<1, 2><<<grp_g / 8, blk, 0, stream>>>(C, W[0], nullptr, nullptr,
                                                G, nullptr, LNN_BS, LNN_DZ, LNN_HID);

  // ---- Jacobian/Hessian/solve: 1 sample per wave, 4 waves/WG ----
  const size_t smem_bytes = (size_t)(4 * (2 * 8192) + 4 * 512 + 4 * (32 * 33)) * 4;
  lnn_jac_hess<<<LNN_BS / 4, 128, smem_bytes, stream>>>(z, W[0], W[1], W[2], W[3],
                                                        U, S, G, out);
}